// FermiNet_31997506355402
// MI455X (gfx1250) — compile-verified
//
#include <hip/hip_runtime.h>
#include <math.h>

// ---------------------------------------------------------------------------
// FermiNet forward, MI455X (gfx1250, wave32).
// Dominant cost: streaming v_matrices (~304MB unique) -> HBM-bandwidth bound
// (AI ~ 0.6 flop/byte; floor ~ 304MB / 23.3TB/s ~ 13us).
// Per-electron GEMV and final fd GEMM use V_WMMA_F32_16X16X4_F32.
// GEMV is K-sliced 4x (4096 waves) for memory-level parallelism; stream is
// prefetched (global_prefetch_b8) with a wave-uniform guard; hot loop uses
// running pointers so all loads use immediate offsets.
// ---------------------------------------------------------------------------

typedef float v2f __attribute__((ext_vector_type(2)));
typedef float v8f __attribute__((ext_vector_type(8)));

#define N_EL  32
#define NUP   16
#define NI    128
#define NKDET 16
#define HS    512
#define HD    4
#define FLEN  1544      // 3*HS + 2*HD
#define NCHUNK (FLEN/4) // 386 K=4 chunks
#define NSLICE 4
#define CPS   97        // ceil(386/4) chunks per slice
#define PFROW 64        // prefetch distance in V rows (~128KB ahead)

// ---------------------------------------------------------------------------
// Phase 0: geometry -> sh0, dh0, eN_norm.  grid: 32 x 128
// ---------------------------------------------------------------------------
__global__ void k_setup(const float* __restrict__ epos,
                        const float* __restrict__ npos,
                        float* __restrict__ enorm,
                        float* __restrict__ sh,
                        float* __restrict__ dh) {
    const int n = blockIdx.x;
    const int i = threadIdx.x;
    const float ex = epos[n*3+0], ey = epos[n*3+1], ez = epos[n*3+2];
    const float dx = ex - npos[i*3+0];
    const float dy = ey - npos[i*3+1];
    const float dz = ez - npos[i*3+2];
    const float r  = sqrtf(dx*dx + dy*dy + dz*dz);
    enorm[n*NI + i]     = r;
    sh[n*HS + 3*i + 0]  = dx;
    sh[n*HS + 3*i + 1]  = dy;
    sh[n*HS + 3*i + 2]  = dz;
    sh[n*HS + 3*NI + i] = r;
    if (i < N_EL) {
        const int m = i;
        const float gx = ex - epos[m*3+0];
        const float gy = ey - epos[m*3+1];
        const float gz = ez - epos[m*3+2];
        const float rr = sqrtf(gx*gx + gy*gy + gz*gz);
        float* p = dh + (n*N_EL + m)*HD;
        p[0] = gx; p[1] = gy; p[2] = gz; p[3] = rr;
    }
}

// ---------------------------------------------------------------------------
// g_up / g_dn means.  g[0..511]=g_up, g[512..1023]=g_dn
// ---------------------------------------------------------------------------
__global__ void k_gmean(const float* __restrict__ sh, float* __restrict__ g) {
    const int o = blockIdx.x*blockDim.x + threadIdx.x;
    if (o >= HS) return;
    float su = 0.f, sd = 0.f;
    for (int n = 0;   n < NUP;  ++n) su += sh[n*HS + o];
    for (int n = NUP; n < N_EL; ++n) sd += sh[n*HS + o];
    g[o]      = su * (1.f/NUP);
    g[HS + o] = sd * (1.f/(N_EL-NUP));
}

// ---------------------------------------------------------------------------
// f[n] = [sh_n | g_up | g_dn | dg_up_n | dg_dn_n].  grid: 32 x 256
// ---------------------------------------------------------------------------
__global__ void k_feat(const float* __restrict__ sh,
                       const float* __restrict__ dh,
                       const float* __restrict__ g,
                       float* __restrict__ f) {
    const int n = blockIdx.x;
    float* fn = f + n*FLEN;
    for (int idx = threadIdx.x; idx < FLEN; idx += blockDim.x) {
        float v;
        if (idx < HS) {
            v = sh[n*HS + idx];
        } else if (idx < 3*HS) {
            v = g[idx - HS];               // g_up then g_dn, contiguous
        } else {
            int j = idx - 3*HS;
            float s = 0.f;
            if (j < HD) {
                for (int m = 0; m < NUP; ++m) s += dh[(n*N_EL+m)*HD + j];
                v = s * (1.f/NUP);
            } else {
                j -= HD;
                for (int m = NUP; m < N_EL; ++m) s += dh[(n*N_EL+m)*HD + j];
                v = s * (1.f/(N_EL-NUP));
            }
        }
        fn[idx] = v;
    }
}

// ---------------------------------------------------------------------------
// K-sliced per-electron GEMV via WMMA rank-4 updates.
// part[s][n][o] = sum_{f in slice s} V[n][f][o] * x[n][f]
// One wave per (o-tile, electron, slice): 32 x 32 x 4 = 4096 waves.
// A (16x4): x-chunk broadcast down M (lane<16 -> K={0,1}; else K={2,3}).
// B (4x16): native row-major V tile.
// D rows identical -> lane reads c[0] = partial for column (lane&15).
// Hot loop: running pointers, all loads/prefetches at immediate offsets.
// ---------------------------------------------------------------------------
__global__ void __launch_bounds__(32)
k_stream_wmma(const float* __restrict__ Vl,
              const float* __restrict__ f,
              float* __restrict__ part) {
    const int n    = blockIdx.y;
    const int o0   = blockIdx.x * 16;
    const int sl   = blockIdx.z;
    const int lane = threadIdx.x;
    const int col  = lane & 15;
    const int kr   = (lane >> 4) * 2;          // 0 or 2

    const float* __restrict__ Vn = Vl + (size_t)n * (FLEN*HS);
    const float* __restrict__ fn = f + n*FLEN;

    const int start = sl * CPS;
    const int end   = (start + CPS < NCHUNK) ? (start + CPS) : NCHUNK;

    // running pointers: row (start*4 + kr), column (o0 + col)
    const float* vptr = Vn + (size_t)(start*4 + kr)*HS + o0 + col;
    const float* fptr = fn + start*4 + kr;

    v8f c0 = {}; v8f c1 = {};
    const int pairs = (end - start) >> 1;
    int row0 = start*4;                        // wave-uniform row of chunk 0
    #pragma unroll 1
    for (int it = 0; it < pairs; ++it) {
        // wave-uniform prefetch guard: kr <= 2, so +3 bounds all lanes
        if (row0 + PFROW + 3 < FLEN) {
            __builtin_prefetch(vptr + (size_t)PFROW*HS,     0, 1);
            __builtin_prefetch(vptr + (size_t)(PFROW+1)*HS, 0, 1);
        }
        v2f a0, b0;
        a0[0] = fptr[0];
        a0[1] = fptr[1];
        b0[0] = vptr[0];
        b0[1] = vptr[HS];
        c0 = __builtin_amdgcn_wmma_f32_16x16x4_f32(false, a0, false, b0,
                                                   (short)0, c0, false, false);
        v2f a1, b1;
        a1[0] = fptr[4];
        a1[1] = fptr[5];
        b1[0] = vptr[4*HS];
        b1[1] = vptr[5*HS];
        c1 = __builtin_amdgcn_wmma_f32_16x16x4_f32(false, a1, false, b1,
                                                   (short)0, c1, false, false);
        vptr += 8*HS;
        fptr += 8;
        row0 += 8;
    }
    if ((end - start) & 1) {                   // odd tail chunk
        v2f a0, b0;
        a0[0] = fptr[0];
        a0[1] = fptr[1];
        b0[0] = vptr[0];
        b0[1] = vptr[HS];
        c0 = __builtin_amdgcn_wmma_f32_16x16x4_f32(false, a0, false, b0,
                                                   (short)0, c0, false, false);
    }
    const float val = c0[0] + c1[0];
    if (lane < 16) {
        part[(size_t)sl*(N_EL*HS) + n*HS + o0 + col] = val;
    }
}

// ---------------------------------------------------------------------------
// Reduce the 4 K-slice partials, apply bias + tanh + residual.
// ---------------------------------------------------------------------------
__global__ void k_reduce(const float* __restrict__ part,
                         const float* __restrict__ bl,
                         const float* __restrict__ sh_in,
                         float* __restrict__ sh_out) {
    const int t = blockIdx.x*blockDim.x + threadIdx.x;   // 0..16383
    if (t >= N_EL*HS) return;
    float s = part[t];
    #pragma unroll
    for (int sl = 1; sl < NSLICE; ++sl) s += part[sl*(N_EL*HS) + t];
    sh_out[t] = tanhf(s + bl[t]) + sh_in[t];
}

// ---------------------------------------------------------------------------
// Pair-stream: 4x4 matvec + tanh + residual per (m,n).
// ---------------------------------------------------------------------------
__global__ void k_pair(const float* __restrict__ Wl,
                       const float* __restrict__ Cl,
                       const float* __restrict__ dh_in,
                       float* __restrict__ dh_out) {
    const int t = blockIdx.x*blockDim.x + threadIdx.x;
    if (t >= N_EL*N_EL) return;
    const float* h = dh_in + t*HD;
    const float* w = Wl + t*HD*HD;
    const float* c = Cl + t*HD;
    float hv[HD];
    #pragma unroll
    for (int j = 0; j < HD; ++j) hv[j] = h[j];
    float* d = dh_out + t*HD;
    #pragma unroll
    for (int i = 0; i < HD; ++i) {
        float s = c[i];
        #pragma unroll
        for (int j = 0; j < HD; ++j) s += w[i*HD + j]*hv[j];
        d[i] = tanhf(s) + hv[i];
    }
}

// ---------------------------------------------------------------------------
// fd[k] = FW[k](32x512) . sh^T(512x32) + bias : full-efficiency WMMA GEMM.
// grid: (4 tiles, 16 k), one wave each; 128 K-chunks, 2 accumulators.
// ---------------------------------------------------------------------------
__global__ void __launch_bounds__(32)
k_fd_wmma(const float* __restrict__ FW,
          const float* __restrict__ FB,
          const float* __restrict__ sh,
          float* __restrict__ fd) {
    const int k    = blockIdx.y;
    const int i0   = (blockIdx.x >> 1) * 16;
    const int j0   = (blockIdx.x & 1) * 16;
    const int lane = threadIdx.x;
    const int row  = lane & 15;                // A row / B col
    const int kb   = (lane >> 4) * 2;

    const float* __restrict__ Ap = FW + (size_t)k*N_EL*HS + (i0+row)*HS + kb;
    const float* __restrict__ Bp = sh + (size_t)(j0+row)*HS + kb;

    v8f c0 = {}, c1 = {};
    #pragma unroll 1
    for (int d0 = 0; d0 < HS; d0 += 8) {
        v2f a0, b0;
        a0[0] = Ap[0];
        a0[1] = Ap[1];
        b0[0] = Bp[0];
        b0[1] = Bp[1];
        c0 = __builtin_amdgcn_wmma_f32_16x16x4_f32(false, a0, false, b0,
                                                   (short)0, c0, false, false);
        v2f a1, b1;
        a1[0] = Ap[4];
        a1[1] = Ap[5];
        b1[0] = Bp[4];
        b1[1] = Bp[5];
        c1 = __builtin_amdgcn_wmma_f32_16x16x4_f32(false, a1, false, b1,
                                                   (short)0, c1, false, false);
        Ap += 8;
        Bp += 8;
    }
    // C/D layout: VGPR r -> M=r (lanes 0-15) / M=r+8 (lanes 16-31), N=lane&15
    #pragma unroll
    for (int r = 0; r < 8; ++r) {
        const int i = i0 + r + (lane >> 4)*8;
        const int j = j0 + (lane & 15);
        fd[(size_t)k*N_EL*N_EL + i*N_EL + j] = c0[r] + c1[r] + FB[k*N_EL + i];
    }
}

// ---------------------------------------------------------------------------
// phi[k,i,j] = fd[k,i,j] * sum_m pi[k,i,m] * exp(-|sigma[k,i,m]|*eN_norm[j,m])
// ---------------------------------------------------------------------------
__global__ void k_env(const float* __restrict__ PI,
                      const float* __restrict__ SG,
                      const float* __restrict__ enorm,
                      const float* __restrict__ fd,
                      float* __restrict__ phi) {
    const int t = blockIdx.x*blockDim.x + threadIdx.x;
    if (t >= NKDET*N_EL*N_EL) return;
    const int k = t >> 10;
    const int i = (t >> 5) & 31;
    const int j = t & 31;
    const float* pk = PI + ((size_t)k*N_EL + i)*NI;
    const float* sk = SG + ((size_t)k*N_EL + i)*NI;
    const float* en = enorm + j*NI;
    float s = 0.f;
    for (int m = 0; m < NI; ++m)
        s += pk[m] * expf(-fabsf(sk[m]) * en[m]);
    phi[t] = fd[t] * s;
}

// ---------------------------------------------------------------------------
// 32 determinants of 16x16 (per-thread LU w/ partial pivoting in LDS),
// then out = sum_k omega[k]*det_up[k]*det_dn[k].
// ---------------------------------------------------------------------------
__global__ void __launch_bounds__(32)
k_det(const float* __restrict__ phi,
      const float* __restrict__ omega,
      float* __restrict__ out) {
    __shared__ float Msh[32][256];
    __shared__ float dets[32];
    const int t = threadIdx.x;
    const int k = t >> 1, half = t & 1;
    const float* pk = phi + (size_t)k*N_EL*N_EL + half*(16*N_EL + 16);
    float* m = Msh[t];
    for (int r = 0; r < 16; ++r)
        for (int c = 0; c < 16; ++c)
            m[r*16 + c] = pk[r*N_EL + c];
    float det = 1.f;
    for (int p = 0; p < 16; ++p) {
        int piv = p; float mx = fabsf(m[p*16 + p]);
        for (int r = p+1; r < 16; ++r) {
            const float v = fabsf(m[r*16 + p]);
            if (v > mx) { mx = v; piv = r; }
        }
        if (piv != p) {
            det = -det;
            for (int c = p; c < 16; ++c) {
                const float tmp = m[p*16 + c];
                m[p*16 + c] = m[piv*16 + c];
                m[piv*16 + c] = tmp;
            }
        }
        const float d = m[p*16 + p];
        det *= d;
        if (d != 0.f) {
            const float inv = 1.f/d;
            for (int r = p+1; r < 16; ++r) {
                const float fct = m[r*16 + p]*inv;
                for (int c = p+1; c < 16; ++c)
                    m[r*16 + c] -= fct*m[p*16 + c];
            }
        }
    }
    dets[t] = det;
    __syncthreads();
    if (t == 0) {
        float s = 0.f;
        for (int kk = 0; kk < NKDET; ++kk)
            s += omega[kk]*dets[2*kk]*dets[2*kk+1];
        out[0] = s;
    }
}

// ---------------------------------------------------------------------------
extern "C" void kernel_launch(void* const* d_in, const int* in_sizes, int n_in,
                              void* d_out, int out_size, void* d_ws, size_t ws_size,
                              hipStream_t stream) {
    (void)in_sizes; (void)n_in; (void)out_size; (void)ws_size;
    const float* epos = (const float*)d_in[0];
    const float* npos = (const float*)d_in[1];
    const float* Vm   = (const float*)d_in[2];   // [4,32,1544,512]
    const float* Bv   = (const float*)d_in[3];   // [4,32,512]
    const float* Wm   = (const float*)d_in[4];   // [4,32,32,4,4]
    const float* Cv   = (const float*)d_in[5];   // [4,32,32,4]
    const float* FW   = (const float*)d_in[6];   // [16,32,512]
    const float* FB   = (const float*)d_in[7];   // [16,32]
    const float* PIw  = (const float*)d_in[8];   // [16,32,128]
    const float* SGw  = (const float*)d_in[9];   // [16,32,128]
    const float* OM   = (const float*)d_in[10];  // [16]
    float* out = (float*)d_out;

    float* ws    = (float*)d_ws;
    float* enorm = ws;                 // 32*128            = 4096
    float* shA   = enorm + 4096;       // 32*512            = 16384
    float* shB   = shA   + 16384;
    float* dhA   = shB   + 16384;      // 32*32*4           = 4096
    float* dhB   = dhA   + 4096;
    float* feat  = dhB   + 4096;       // 32*1544           = 49408
    float* g     = feat  + 49408;      // 1024
    float* fd    = g     + 1024;       // 16*32*32          = 16384
    float* phi   = fd    + 16384;      // 16384
    float* part  = phi   + 16384;      // 4*32*512          = 65536  (~1MB tot)

    k_setup<<<N_EL, NI, 0, stream>>>(epos, npos, enorm, shA, dhA);

    const int order[4] = {0, 1, 2, 2};   // reference quirk: layers 0,1,2,2
    float* sc = shA; float* sn = shB;
    float* dc = dhA; float* dn = dhB;
    for (int a = 0; a < 4; ++a) {
        const int li = order[a];
        k_gmean<<<2, 256, 0, stream>>>(sc, g);
        k_feat <<<N_EL, 256, 0, stream>>>(sc, dc, g, feat);
        k_stream_wmma<<<dim3(HS/16, N_EL, NSLICE), 32, 0, stream>>>(
            Vm + (size_t)li*N_EL*FLEN*HS, feat, part);
        k_reduce<<<(N_EL*HS)/256, 256, 0, stream>>>(
            part, Bv + (size_t)li*N_EL*HS, sc, sn);
        k_pair<<<4, 256, 0, stream>>>(
            Wm + (size_t)li*N_EL*N_EL*HD*HD,
            Cv + (size_t)li*N_EL*N_EL*HD,
            dc, dn);
        float* tp;
        tp = sc; sc = sn; sn = tp;
        tp = dc; dc = dn; dn = tp;
    }

    k_fd_wmma<<<dim3(4, NKDET), 32, 0, stream>>>(FW, FB, sc, fd);
    k_env<<<NKDET*N_EL*N_EL/256, 256, 0, stream>>>(PIw, SGw, enorm, fd, phi);
    k_det<<<1, 32, 0, stream>>>(phi, OM, out);
}